// NonLinearActorNet_8873402434288
// MI455X (gfx1250) — compile-verified
//
#include <hip/hip_runtime.h>
#include <hip/hip_bf16.h>

// ---------------------------------------------------------------------------
// NonLinearActorNet for MI455X (gfx1250, wave32, WMMA).
//
// Sizes (fixed by reference): B=1024, D=512, H=512, A=18, T=64.
// FC1 (537 MFLOP) is done as per-task GEMMs via V_WMMA_F32_16X16X4_F32 after a
// deterministic counting sort groups samples by task (64 tasks -> W1 fits L2).
// FC2 + log-softmax + exact JAX Threefry/Gumbel categorical sample done per
// sample by one wave.
// ---------------------------------------------------------------------------

typedef __attribute__((ext_vector_type(2))) float v2f;
typedef __attribute__((ext_vector_type(8))) float v8f;

#define NB 1024
#define ND 512
#define NH 512
#define NA 18
#define NT 64

// ---------------------------------------------------------------------------
// Kernel 1: deterministic grouping of samples by task (counting sort).
// One thread per task; no atomics => order[] is bit-stable across calls.
// ---------------------------------------------------------------------------
__global__ __launch_bounds__(64) void group_kernel(const int* __restrict__ task_id,
                                                   int* __restrict__ counts,
                                                   int* __restrict__ starts,
                                                   int* __restrict__ order) {
  __shared__ int s_cnt[NT];
  __shared__ int s_start[NT];
  const int t = threadIdx.x;
  int c = 0;
  for (int i = 0; i < NB; ++i) c += (task_id[i] == t);
  s_cnt[t] = c;
  __syncthreads();
  if (t == 0) {
    int acc = 0;
    for (int j = 0; j < NT; ++j) { s_start[j] = acc; acc += s_cnt[j]; }
  }
  __syncthreads();
  int pos = s_start[t];
  for (int i = 0; i < NB; ++i)
    if (task_id[i] == t) order[pos++] = i;
  counts[t] = s_cnt[t];
  starts[t] = s_start[t];
}

// ---------------------------------------------------------------------------
// Kernel 2: FC1 as per-task GEMM tiles with V_WMMA_F32_16X16X4_F32.
// Block = 128 threads (4 waves). Each block computes a 16(sample) x 64(H) tile
// for one task. Wave w owns 16 consecutive H columns.
// Grid: x = H/64 tiles, y = worst-case row tiles (64), z = task. Early exit.
// ---------------------------------------------------------------------------
__global__ __launch_bounds__(128) void fc1_kernel(const float* __restrict__ xs,
                                                  const float* __restrict__ w1,
                                                  const float* __restrict__ b1,
                                                  const int* __restrict__ order,
                                                  const int* __restrict__ starts,
                                                  const int* __restrict__ counts,
                                                  float* __restrict__ h) {
  const int t  = blockIdx.z;
  const int rt = blockIdx.y;           // 16-sample row tile within task group
  const int nb = blockIdx.x;           // 64-column tile of H
  const int cnt = counts[t];
  if (rt * 16 >= cnt) return;          // over-provisioned grid: early exit
  const int start = starts[t];

  // LDS staging. Xs stored transposed [k][m] so A-fragment reads by lanes
  // 0..15 (distinct m, same k) are consecutive (no bank conflicts).
  // Ws row padded to 68 floats so the lane16..31 half (k+2) shifts banks.
  __shared__ float Xs[64][16];
  __shared__ float Ws[64][68];

  const int tid  = threadIdx.x;
  const int lane = tid & 31;
  const int wave = tid >> 5;
  const int hi   = (lane >= 16) ? 1 : 0;
  const int mi   = lane & 15;          // A fragment: M index of this lane
  const int ni   = lane & 15;          // B fragment: N index of this lane
  const int n0   = wave * 16;          // wave's N sub-tile inside the 64 cols

  v8f acc = {};                        // 16x16 f32 C/D accumulator (8 VGPRs)

  for (int k0 = 0; k0 < ND; k0 += 64) {
    // Stage X tile: 16 samples x 64 k. Coalesced reads (kk fastest).
    for (int e = tid; e < 16 * 64; e += 128) {
      const int m  = e >> 6;
      const int kk = e & 63;
      int r = rt * 16 + m;
      r = (r < cnt) ? r : (cnt - 1);   // pad tile by duplicating last row
      const int s = order[start + r];
      Xs[kk][m] = xs[s * ND + k0 + kk];
    }
    // Stage W1 tile: 64 k x 64 n. Coalesced reads (nn fastest).
    for (int e = tid; e < 64 * 64; e += 128) {
      const int kk = e >> 6;
      const int nn = e & 63;
      Ws[kk][nn] = w1[((size_t)t * ND + (k0 + kk)) * NH + nb * 64 + nn];
    }
    __syncthreads();

    // 16 WMMA steps of K=4 each.
    for (int kk = 0; kk < 64; kk += 4) {
      const int koff = kk + (hi ? 2 : 0);
      v2f a, b;
      // A 16x4 f32: VGPR0 = K (0|2), VGPR1 = K (1|3) per lane-half.
      a.x = Xs[koff + 0][mi];
      a.y = Xs[koff + 1][mi];
      // B 4x16 f32: symmetric layout, lane holds N=ni, VGPRp holds row K.
      b.x = Ws[koff + 0][n0 + ni];
      b.y = Ws[koff + 1][n0 + ni];
      acc = __builtin_amdgcn_wmma_f32_16x16x4_f32(
          /*neg_a=*/false, a, /*neg_b=*/false, b,
          /*c_mod=*/(short)0, acc, /*reuse_a=*/false, /*reuse_b=*/false);
    }
    __syncthreads();
  }

  // Epilogue: bias + ReLU, scatter back to original sample index.
  const int ncol = nb * 64 + n0 + ni;
  const float bias = b1[t * NH + ncol];
  for (int v = 0; v < 8; ++v) {
    const int m = v + (hi ? 8 : 0);    // C/D layout: VGPR v -> M=v / M=v+8
    const int r = rt * 16 + m;
    if (r < cnt) {
      const int s = order[start + r];
      const float val = acc[v] + bias;
      h[(size_t)s * NH + ncol] = val > 0.0f ? val : 0.0f;
    }
  }
}

// ---------------------------------------------------------------------------
// Exact JAX Threefry-2x32 (key = jax.random.key(1) -> (0,1)).
// ---------------------------------------------------------------------------
__device__ __forceinline__ unsigned rotl32(unsigned x, int d) {
  return (x << d) | (x >> (32 - d));
}

__device__ void threefry2x32(unsigned k0, unsigned k1, unsigned c0, unsigned c1,
                             unsigned* o0, unsigned* o1) {
  const unsigned ks2 = k0 ^ k1 ^ 0x1BD11BDAu;
  unsigned x0 = c0 + k0, x1 = c1 + k1;
  const int r1[4] = {13, 15, 26, 6};
  const int r2[4] = {17, 29, 16, 24};
#pragma unroll
  for (int i = 0; i < 4; ++i) { x0 += x1; x1 = rotl32(x1, r1[i]); x1 ^= x0; }
  x0 += k1; x1 += ks2 + 1u;
#pragma unroll
  for (int i = 0; i < 4; ++i) { x0 += x1; x1 = rotl32(x1, r2[i]); x1 ^= x0; }
  x0 += ks2; x1 += k0 + 2u;
#pragma unroll
  for (int i = 0; i < 4; ++i) { x0 += x1; x1 = rotl32(x1, r1[i]); x1 ^= x0; }
  x0 += k0; x1 += k1 + 3u;
#pragma unroll
  for (int i = 0; i < 4; ++i) { x0 += x1; x1 = rotl32(x1, r2[i]); x1 ^= x0; }
  x0 += k1; x1 += ks2 + 4u;
#pragma unroll
  for (int i = 0; i < 4; ++i) { x0 += x1; x1 = rotl32(x1, r1[i]); x1 ^= x0; }
  x0 += ks2; x1 += k0 + 5u;
  *o0 = x0; *o1 = x1;
}

// random bits for flat index i of an N=B*A draw (split-counter layout used by
// jax's threefry_2x32: x = [iota[:N/2], iota[N/2:]]).
__device__ unsigned jax_random_bits(unsigned i) {
  const unsigned half = (NB * NA) / 2;  // 9216
  unsigned o0, o1;
  if (i < half) { threefry2x32(0u, 1u, i, i + half, &o0, &o1); return o0; }
  threefry2x32(0u, 1u, i - half, i, &o0, &o1);
  return o1;
}

// ---------------------------------------------------------------------------
// Kernel 3: FC2 + log_softmax + categorical(key(1)) + log_prob + entropy.
// One wave (32 threads) per sample; lanes 0..17 compute logits.
// ---------------------------------------------------------------------------
__global__ __launch_bounds__(32) void fc2_kernel(const float* __restrict__ h,
                                                 const int* __restrict__ task_id,
                                                 const float* __restrict__ w2,
                                                 const float* __restrict__ b2,
                                                 float* __restrict__ out) {
  const int b = blockIdx.x;
  const int lane = threadIdx.x;
  const int t = task_id[b];
  __shared__ float logits[NA];

  if (lane < NA) {
    float acc = b2[t * NA + lane];
    const float* hb = h + (size_t)b * NH;
    const float* w = w2 + (size_t)t * NH * NA + lane;
    for (int k = 0; k < NH; ++k) acc = fmaf(hb[k], w[k * NA], acc);
    logits[lane] = acc;
  }
  __syncthreads();

  if (lane == 0) {
    // log-softmax
    float mx = logits[0];
    for (int a = 1; a < NA; ++a) mx = fmaxf(mx, logits[a]);
    float se = 0.0f;
    for (int a = 0; a < NA; ++a) se += __expf(logits[a] - mx);
    const float lse = mx + __logf(se);

    // Gumbel-max sample, matching jax.random.categorical(key(1), logits).
    const float tiny = 1.1754943508222875e-38f;
    int best = 0;
    float bestv = -3.402823466e+38f;
    for (int a = 0; a < NA; ++a) {
      const unsigned bits = jax_random_bits((unsigned)(b * NA + a));
      float f = __uint_as_float((bits >> 9) | 0x3f800000u) - 1.0f;  // [0,1)
      float u = fmaxf(f * (1.0f - tiny) + tiny, tiny);              // [tiny,1)
      const float g = -__logf(-__logf(u));
      const float v = logits[a] + g;
      if (v > bestv) { bestv = v; best = a; }  // strict '>' => first max wins
    }

    float ent = 0.0f;
    for (int a = 0; a < NA; ++a) {
      const float lp = logits[a] - lse;
      ent -= __expf(lp) * lp;
    }

    out[b]            = (float)best;           // action
    out[NB + b]       = logits[best] - lse;    // log_prob [B,1]
    out[2 * NB + b]   = ent;                   // entropy  [B,1]
  }
}

// ---------------------------------------------------------------------------
// Launch. Workspace layout (needs ~2.01 MB of d_ws):
//   [0,256)      int counts[64]
//   [256,512)    int starts[64]
//   [512,4608)   int order[1024]
//   [8192, +2MB) float h[1024*512]
// ---------------------------------------------------------------------------
extern "C" void kernel_launch(void* const* d_in, const int* in_sizes, int n_in,
                              void* d_out, int out_size, void* d_ws, size_t ws_size,
                              hipStream_t stream) {
  const float* xs      = (const float*)d_in[0];
  const int*   task_id = (const int*)d_in[1];   // jax default x64 off -> int32
  const float* w1      = (const float*)d_in[2];
  const float* b1      = (const float*)d_in[3];
  const float* w2      = (const float*)d_in[4];
  const float* b2      = (const float*)d_in[5];
  float* out = (float*)d_out;

  char* ws = (char*)d_ws;
  int*   counts = (int*)(ws);
  int*   starts = (int*)(ws + 256);
  int*   order  = (int*)(ws + 512);
  float* h      = (float*)(ws + 8192);

  group_kernel<<<dim3(1), dim3(64), 0, stream>>>(task_id, counts, starts, order);

  dim3 g2(NH / 64, NB / 16, NT);  // (8, 64, 64); most (y,z) blocks early-exit
  fc1_kernel<<<g2, dim3(128), 0, stream>>>(xs, w1, b1, order, starts, counts, h);

  fc2_kernel<<<dim3(NB), dim3(32), 0, stream>>>(h, task_id, w2, b2, out);
}